// PostProcessor_23055384445238
// MI455X (gfx1250) — compile-verified
//
#include <hip/hip_runtime.h>
#include <math.h>

#define B_    64
#define C_    3
#define H_    96
#define W_    320
#define HW_   (H_*W_)
#define TOPK_ 50
#define NSLICE (B_*C_)
#define PI_F  3.14159265f
#define NEGINF (-3.0e38f)
#define TILE_ 8                  // rows of NMS computed per staging step
#define BROWS_ (TILE_ + 2)       // staged rows (with halo)

typedef float v2f __attribute__((ext_vector_type(2)));
typedef float v8f __attribute__((ext_vector_type(8)));
typedef int   v4i_ __attribute__((vector_size(16)));   // matches builtin param pointee
typedef __attribute__((address_space(1))) v4i_* gbl4_t;
typedef __attribute__((address_space(3))) v4i_* lds4_t;

#if defined(__AMDGCN__) && __has_builtin(__builtin_amdgcn_global_load_async_to_lds_b128)
#define HAVE_ASYNC 1
#else
#define HAVE_ASYNC 0
#endif

#if HAVE_ASYNC
// 16B global -> LDS async copy (ASYNCcnt-tracked, no VGPR staging)
__device__ inline void async_copy16(const float* g, float* l) {
  __builtin_amdgcn_global_load_async_to_lds_b128(
      (gbl4_t)(unsigned long long)(uintptr_t)g,
      (lds4_t)(unsigned)(uintptr_t)l,
      0, 0);
}
__device__ inline void wait_async0() {
#if __has_builtin(__builtin_amdgcn_s_wait_asynccnt)
  __builtin_amdgcn_s_wait_asynccnt(0);
#else
  asm volatile("s_wait_asynccnt 0x0" ::: "memory");
#endif
}
#endif

// ---------------------------------------------------------------------------
// Kernel A: 3x3 NMS + exact per-(b,c) top-50.
// One workgroup per (b,c) slice. Heat rows are staged band-by-band into LDS
// (async data-mover path when available), NMS result parked in 120KB of LDS,
// then a hierarchical argmax: per-thread chunk maxima cached in registers,
// only the winning thread rescans its 120-element chunk each round.
// ---------------------------------------------------------------------------
__global__ __launch_bounds__(256) void nms_topk_kernel(
    const float* __restrict__ heat,
    float* __restrict__ ws_vals, int* __restrict__ ws_inds)
{
  __shared__ float s_nms[HW_];            // 120 KB
  __shared__ float s_raw[BROWS_ * W_];    // 12.8 KB staging band
  __shared__ float s_rv[256];
  __shared__ int   s_ri[256];

  const int slice = blockIdx.x;                 // b*C + c
  const float* hp = heat + (size_t)slice * HW_;
  const int tid = threadIdx.x;

  // Phase 1: staged NMS. Replicate-clamp halo rows == SAME -inf pad for max.
  for (int s = 0; s < H_ / TILE_; ++s) {
    const int r0     = s * TILE_;
    const int gfirst = (r0 == 0) ? 0 : r0 - 1;
    const int glast  = (r0 + TILE_ >= H_) ? H_ - 1 : r0 + TILE_;
    const int nch    = (glast - gfirst + 1) * (W_ / 4);
    const int bstart = (r0 == 0) ? 1 : 0;       // global row 'gfirst' -> this buf row
    const float* src = hp + gfirst * W_;
    float* dst = &s_raw[bstart * W_];

#if HAVE_ASYNC
    for (int c = tid; c < nch; c += 256) async_copy16(src + c * 4, dst + c * 4);
    if (r0 == 0)                                  // duplicate top row into halo
      for (int c = tid; c < W_ / 4; c += 256)
        async_copy16(hp + c * 4, &s_raw[c * 4]);
    if (r0 + TILE_ >= H_)                         // duplicate bottom row into halo
      for (int c = tid; c < W_ / 4; c += 256)
        async_copy16(hp + (H_ - 1) * W_ + c * 4, &s_raw[(TILE_ + 1) * W_ + c * 4]);
    wait_async0();
#else
    for (int c = tid; c < nch; c += 256)
      *(float4*)(dst + c * 4) = *(const float4*)(src + c * 4);
    if (r0 == 0)
      for (int c = tid; c < W_ / 4; c += 256)
        *(float4*)(&s_raw[c * 4]) = *(const float4*)(hp + c * 4);
    if (r0 + TILE_ >= H_)
      for (int c = tid; c < W_ / 4; c += 256)
        *(float4*)(&s_raw[(TILE_ + 1) * W_ + c * 4]) =
            *(const float4*)(hp + (H_ - 1) * W_ + c * 4);
#endif
    __syncthreads();

    // global row r0+lr always lives at buffer row lr+1
    for (int i = tid; i < TILE_ * W_; i += 256) {
      int lr = i / W_, x = i - lr * W_;
      int x0 = (x > 0) ? x - 1 : 0, x1 = (x < W_ - 1) ? x + 1 : W_ - 1;
      float v = s_raw[(lr + 1) * W_ + x];
      float m = v;
      #pragma unroll
      for (int br = 0; br < 3; ++br)
        for (int xx = x0; xx <= x1; ++xx) {
          float n = s_raw[(lr + br) * W_ + xx];
          m = (n > m) ? n : m;
        }
      s_nms[(r0 + lr) * W_ + x] = (v == m) ? v : 0.0f;  // heat * (hmax == heat)
    }
    __syncthreads();
  }

  // Phase 2: hierarchical top-50 (value desc, index asc ties = lax.top_k).
  const int CH = HW_ / 256;            // 120-element chunk per thread
  const int cbase = tid * CH;
  float myv = NEGINF; int myi = cbase;
  for (int j = 0; j < CH; ++j) {       // ascending scan: '>' keeps lowest idx
    float v = s_nms[cbase + j];
    if (v > myv) { myv = v; myi = cbase + j; }
  }

  for (int k = 0; k < TOPK_; ++k) {
    s_rv[tid] = myv; s_ri[tid] = myi;
    __syncthreads();
    for (int off = 128; off > 0; off >>= 1) {
      if (tid < off) {
        float ov = s_rv[tid + off]; int oi = s_ri[tid + off];
        if (ov > s_rv[tid] || (ov == s_rv[tid] && oi < s_ri[tid])) {
          s_rv[tid] = ov; s_ri[tid] = oi;
        }
      }
      __syncthreads();
    }
    float wv = s_rv[0]; int wi = s_ri[0];
    if (tid == 0) {
      ws_vals[slice * TOPK_ + k] = wv;
      ws_inds[slice * TOPK_ + k] = wi;
    }
    if (wi >= cbase && wi < cbase + CH) {   // only the owner rescans its chunk
      s_nms[wi] = NEGINF;
      myv = NEGINF; myi = cbase;
      for (int j = 0; j < CH; ++j) {
        float v = s_nms[cbase + j];
        if (v > myv) { myv = v; myi = cbase + j; }
      }
    }
    __syncthreads();
  }
}

// analytic 3x3 inverse
__device__ inline void inv3(const float* m, float* r) {
  float a=m[0],b=m[1],c=m[2],d=m[3],e=m[4],f=m[5],g=m[6],h=m[7],i=m[8];
  float A = e*i - f*h, Bc = -(d*i - f*g), Cc = d*h - e*g;
  float det = a*A + b*Bc + c*Cc;
  float id = 1.0f / det;
  r[0]=A*id;            r[1]=-(b*i-c*h)*id;  r[2]=(b*f-c*e)*id;
  r[3]=Bc*id;           r[4]=(a*i-c*g)*id;   r[5]=-(a*f-c*d)*id;
  r[6]=Cc*id;           r[7]=-(a*h-b*g)*id;  r[8]=(a*e-b*d)*id;
}

__device__ inline float wrap_pi(float a) {
  a = (a >  PI_F) ? a - 2.0f * PI_F : a;
  a = (a < -PI_F) ? a + 2.0f * PI_F : a;
  return a;
}

// ---------------------------------------------------------------------------
// Kernel B: per-batch merge of 3x50 -> top-50, then decode.
// The reference's einsum('nij,nj->ni') 3x3 homogeneous transforms run on
// V_WMMA_F32_16X16X4_F32: A = matrix padded into 16x4, B = 16 points
// [x,y,1,0] per wave, D rows 0..2 = transformed components.
// ---------------------------------------------------------------------------
__global__ __launch_bounds__(64) void decode_kernel(
    const float* __restrict__ reg,      // (B,12,H,W)
    const float* __restrict__ calib,    // (B,3,3)
    const float* __restrict__ trans,    // (B,3,3)
    const float* __restrict__ dimref,   // (3,3)
    const float* __restrict__ ws_vals, const int* __restrict__ ws_inds,
    float* __restrict__ out)            // (B*50, 14)
{
  const int b   = blockIdx.x;
  const int tid = threadIdx.x;

  __shared__ float s_vals[C_ * TOPK_];
  __shared__ int   s_inds[C_ * TOPK_];
  __shared__ float s_rv[64]; __shared__ int s_ri[64];
  __shared__ float sel_score[TOPK_]; __shared__ int sel_ind[TOPK_]; __shared__ int sel_cls[TOPK_];
  __shared__ float Tinv[9], M2[9];
  __shared__ float px[64],  py[64];              // proj points (4 groups of 16)
  __shared__ float qx[128], qy[128];             // lt in [0,64), rb in [64,128)
  __shared__ float pox[64], poy[64], poz[64];    // M2 @ [proj,1]
  __shared__ float qox[128], qoy[128];           // Tinv @ [lt/rb,1] (x,y only)

  for (int i = tid; i < C_ * TOPK_; i += 64) {
    s_vals[i] = ws_vals[b * C_ * TOPK_ + i];
    s_inds[i] = ws_inds[b * C_ * TOPK_ + i];
  }
  __syncthreads();

  // top-50 over 150 (value desc, flattened-index asc, matching lax.top_k)
  for (int k = 0; k < TOPK_; ++k) {
    float bv = NEGINF; int bi = 0;
    for (int i = tid; i < C_ * TOPK_; i += 64) {
      float v = s_vals[i];
      if (v > bv) { bv = v; bi = i; }
    }
    s_rv[tid] = bv; s_ri[tid] = bi;
    __syncthreads();
    for (int off = 32; off > 0; off >>= 1) {
      if (tid < off) {
        float ov = s_rv[tid + off]; int oi = s_ri[tid + off];
        if (ov > s_rv[tid] || (ov == s_rv[tid] && oi < s_ri[tid])) {
          s_rv[tid] = ov; s_ri[tid] = oi;
        }
      }
      __syncthreads();
    }
    if (tid == 0) {
      int p = s_ri[0];
      sel_score[k] = s_rv[0];
      sel_cls[k]   = p / TOPK_;
      sel_ind[k]   = s_inds[p];
      s_vals[p] = NEGINF;
    }
    __syncthreads();
  }

  // per-batch matrices: Tinv = trans^-1, M2 = calib^-1 * trans^-1
  if (tid == 0) {
    float T[9], K[9], Ti[9], Ki[9];
    for (int i = 0; i < 9; ++i) { T[i] = trans[b*9+i]; K[i] = calib[b*9+i]; }
    inv3(T, Ti); inv3(K, Ki);
    for (int i = 0; i < 9; ++i) Tinv[i] = Ti[i];
    for (int r = 0; r < 3; ++r)
      for (int c = 0; c < 3; ++c) {
        float s = 0.f;
        for (int j = 0; j < 3; ++j) s += Ki[r*3+j] * Ti[j*3+c];
        M2[r*3+c] = s;
      }
  }
  __syncthreads();

  // per-detection gather + staging (registers persist across barriers)
  float regv[12]; float ptx = 0.f, pty = 0.f, depth = 0.f, score = 0.f;
  int cls = 0;
  if (tid < TOPK_) {
    int ind = sel_ind[tid]; cls = sel_cls[tid]; score = sel_score[tid];
    int y = ind / W_, x = ind - y * W_;
    const float* rp = reg + (size_t)b * 12 * HW_ + ind;
    #pragma unroll
    for (int ch = 0; ch < 12; ++ch) regv[ch] = rp[(size_t)ch * HW_];
    ptx = (float)x; pty = (float)y;
    depth = regv[0] * 16.32f + 28.01f;
    px[tid] = ptx + regv[1];  py[tid] = pty + regv[2];
    float cx = ptx + regv[8], cy = pty + regv[9];
    qx[tid]      = cx - regv[10] * 0.5f; qy[tid]      = cy - regv[11] * 0.5f; // lt
    qx[64 + tid] = cx + regv[10] * 0.5f; qy[64 + tid] = cy + regv[11] * 0.5f; // rb
  } else {
    px[tid] = 0.f; py[tid] = 0.f;
    qx[tid] = 0.f; qy[tid] = 0.f;
    qx[64 + tid] = 0.f; qy[64 + tid] = 0.f;
  }
  __syncthreads();

  // Wave 0 (EXEC all ones across 32 lanes) runs the WMMA transforms.
  if (tid < 32) {
    const int lane = tid;
    const int mrow = lane & 15;
    const bool hi  = lane >= 16;     // A lanes 16-31 hold K=2,3; B lanes hold [1,0]

    v2f aM2, aT;
    if (!hi) {
      aM2.x = (mrow < 3) ? M2[mrow*3+0]   : 0.f;  aM2.y = (mrow < 3) ? M2[mrow*3+1]   : 0.f;
      aT.x  = (mrow < 3) ? Tinv[mrow*3+0] : 0.f;  aT.y  = (mrow < 3) ? Tinv[mrow*3+1] : 0.f;
    } else {
      aM2.x = (mrow < 3) ? M2[mrow*3+2]   : 0.f;  aM2.y = 0.f;
      aT.x  = (mrow < 3) ? Tinv[mrow*3+2] : 0.f;  aT.y  = 0.f;
    }

    #pragma unroll
    for (int g = 0; g < 4; ++g) {     // loc pre-scale: M2 @ [proj,1]
      v2f bb;
      if (!hi) { bb.x = px[g*16 + mrow]; bb.y = py[g*16 + mrow]; }  // K=0,1
      else     { bb.x = 1.0f;            bb.y = 0.0f;            }  // K=2,3
      v8f c = {};
      v8f d = __builtin_amdgcn_wmma_f32_16x16x4_f32(false, aM2, false, bb,
                                                    (short)0, c, false, false);
      if (!hi) { pox[g*16+mrow] = d[0]; poy[g*16+mrow] = d[1]; poz[g*16+mrow] = d[2]; }
    }
    #pragma unroll
    for (int g = 0; g < 8; ++g) {     // bbox corners: Tinv @ [lt/rb,1]
      v2f bb;
      if (!hi) { bb.x = qx[g*16 + mrow]; bb.y = qy[g*16 + mrow]; }
      else     { bb.x = 1.0f;            bb.y = 0.0f;            }
      v8f c = {};
      v8f d = __builtin_amdgcn_wmma_f32_16x16x4_f32(false, aT, false, bb,
                                                    (short)0, c, false, false);
      if (!hi) { qox[g*16+mrow] = d[0]; qoy[g*16+mrow] = d[1]; }
    }
  }
  __syncthreads();

  // Final per-detection assembly.
  if (tid < TOPK_) {
    float loc0 = pox[tid] * depth, loc1 = poy[tid] * depth, loc2 = poz[tid] * depth;
    float d3a = expf(regv[3]) * dimref[cls*3+0];
    float d3b = expf(regv[4]) * dimref[cls*3+1];
    float d3c = expf(regv[5]) * dimref[cls*3+2];
    loc1 += d3b * 0.5f;
    float rays   = atanf(loc0 / (loc2 + 1e-7f));
    float alphas = atanf(regv[6] / (regv[7] + 1e-7f));
    alphas += (regv[7] >= 0.f) ? (-PI_F * 0.5f) : (PI_F * 0.5f);
    float rotys = wrap_pi(alphas + rays);
    alphas = wrap_pi(alphas);

    float o[14];
    o[0]  = (float)cls;
    o[1]  = alphas;
    o[2]  = qox[tid];       o[3]  = qoy[tid];       // lt -> image
    o[4]  = qox[64 + tid];  o[5]  = qoy[64 + tid];  // rb -> image
    o[6]  = d3b; o[7] = d3c; o[8] = d3a;            // roll(dims3d, -1)
    o[9]  = loc0; o[10] = loc1; o[11] = loc2;
    o[12] = rotys;
    o[13] = score;

    bool keep = score > 0.25f;
    float* op = out + ((size_t)b * TOPK_ + tid) * 14;
    #pragma unroll
    for (int k = 0; k < 14; ++k) op[k] = keep ? o[k] : 0.0f;
  }
}

extern "C" void kernel_launch(void* const* d_in, const int* in_sizes, int n_in,
                              void* d_out, int out_size, void* d_ws, size_t ws_size,
                              hipStream_t stream) {
  const float* heat   = (const float*)d_in[0];
  const float* reg    = (const float*)d_in[1];
  const float* calib  = (const float*)d_in[2];
  const float* trans  = (const float*)d_in[3];
  const float* dimref = (const float*)d_in[4];

  float* ws_vals = (float*)d_ws;
  int*   ws_inds = (int*)((char*)d_ws + (size_t)NSLICE * TOPK_ * sizeof(float));

  nms_topk_kernel<<<NSLICE, 256, 0, stream>>>(heat, ws_vals, ws_inds);
  decode_kernel<<<B_, 64, 0, stream>>>(reg, calib, trans, dimref,
                                       ws_vals, ws_inds, (float*)d_out);
}